// DLO_constrains_loss_2903397892257
// MI455X (gfx1250) — compile-verified
//
#include <hip/hip_runtime.h>

typedef __attribute__((ext_vector_type(2))) float v2f;
typedef __attribute__((ext_vector_type(8))) float v8f;

#define NPTS     131072
#define GN       30
#define GS       100
#define CSTRIDE  4369          // NPTS / GN

// Order-preserving float -> uint key (ascending)
__device__ __forceinline__ unsigned okey(float f) {
  unsigned u = __float_as_uint(f);
  return (u & 0x80000000u) ? ~u : (u | 0x80000000u);
}

// One workgroup = 8 wave32 = 256 threads, handles one (batch, 16-query tile).
// mode 0: write per-query mean (3 floats).  mode 1: write Σx (3) + Σxxᵀ (6).
__global__ __launch_bounds__(256)
void knn_select_kernel(const float* __restrict__ pts,
                       const float* __restrict__ qbase,
                       int qBatchStride, int qPointStride,
                       float* __restrict__ outv, int mode)
{
  __shared__ unsigned hist[16 * 256];      // per-query radix histograms (16 KB)
  __shared__ float    qx[16], qy[16], qz[16];
  __shared__ unsigned prefix[16], kth[16];
  __shared__ float    sums[16][9];
  __shared__ unsigned tiecnt[16];
  __shared__ unsigned tielist[16][16];

  const int tid  = threadIdx.x;
  const int lane = tid & 31;
  const int wave = tid >> 5;
  const int col  = lane & 15;
  const int hlf  = lane >> 4;          // 0: K=0,1 half | 1: K=2,3 half
  const int rbase = hlf * 8;           // D rows: rbase..rbase+7 for this lane

  const int b     = blockIdx.y;
  const int qtile = blockIdx.x;
  const int nq    = (qtile == 0) ? 16 : (GN - 16);

  const float* P = pts + (size_t)b * NPTS * 3;
  const float* Q = qbase + (size_t)b * qBatchStride + (size_t)qtile * 16 * qPointStride;

  if (tid < 16) {
    float x = 0.f, y = 0.f, z = 0.f;
    if (tid < nq) {
      const float* q = Q + (size_t)tid * qPointStride;
      x = q[0]; y = q[1]; z = q[2];
    }
    qx[tid] = x; qy[tid] = y; qz[tid] = z;
    prefix[tid] = 0u; kth[tid] = GS;   // 1-based rank of boundary element
  }
  __syncthreads();

  // Per-lane row-active flags and histogram bases (loop-invariant).
  bool rowAct[8];
  #pragma unroll
  for (int r = 0; r < 8; ++r) rowAct[r] = (rbase + r) < nq;

  // A fragment (16x4 f32): row M = col; lanes 0-15 hold K=0,1; lanes 16-31 K=2,3.
  // A row = [-2qx, -2qy, -2qz, 1]  ->  D = -2 q.p + |p|^2
  v2f afrag;
  if (hlf == 0) { afrag.x = -2.0f * qx[col]; afrag.y = -2.0f * qy[col]; }
  else          { afrag.x = -2.0f * qz[col]; afrag.y = 1.0f;            }

  // ---------- 4 radix-select passes (high byte -> low byte) ----------
  for (int pass = 0; pass < 4; ++pass) {
    for (int i = tid; i < 16 * 256; i += 256) hist[i] = 0u;
    __syncthreads();
    const unsigned shiftD = (unsigned)(24 - 8 * pass);

    // Hoist this pass's per-row prefixes out of the scan loop (stable until
    // the next barrier) so the hot loop has no LDS loads, only ds_add.
    unsigned pf[8];
    #pragma unroll
    for (int r = 0; r < 8; ++r) pf[r] = rowAct[r] ? prefix[rbase + r] : 0u;

    for (int tile = wave; tile < NPTS / 16; tile += 8) {
      const int pidx = tile * 16 + col;
      const float px = P[3 * pidx + 0];
      const float py = P[3 * pidx + 1];
      const float pz = P[3 * pidx + 2];
      const float pn2 = fmaf(px, px, fmaf(py, py, pz * pz));
      v2f bfrag;                       // B col = [px, py, pz, |p|^2]
      if (hlf == 0) { bfrag.x = px; bfrag.y = py;  }
      else          { bfrag.x = pz; bfrag.y = pn2; }
      v8f acc = {};
      acc = __builtin_amdgcn_wmma_f32_16x16x4_f32(false, afrag, false, bfrag,
                                                  (short)0, acc, false, false);
      #pragma unroll
      for (int r = 0; r < 8; ++r) {
        if (rowAct[r]) {
          const unsigned key = okey(acc[r]);
          const bool match = (pass == 0) || ((key >> (shiftD + 8u)) == pf[r]);
          if (match) {
            atomicAdd(&hist[(rbase + r) * 256 + ((key >> shiftD) & 255u)], 1u);
          }
        }
      }
    }
    __syncthreads();

    if (tid < nq) {                    // find byte bin containing the kth element
      unsigned cum = 0u, kr = kth[tid], pfv = prefix[tid];
      for (unsigned dgt = 0; dgt < 256u; ++dgt) {
        const unsigned h = hist[tid * 256 + dgt];
        if (cum + h >= kr) { prefix[tid] = (pfv << 8) | dgt; kth[tid] = kr - cum; break; }
        cum += h;
      }
    }
    __syncthreads();
  }
  // prefix[] = full key T of the 100th smallest; kth[] = #ties at T still needed.

  // ---------- final selection + accumulation ----------
  if (tid < 16) tiecnt[tid] = 0u;
  for (int i = tid; i < 16 * 9; i += 256) ((float*)sums)[i] = 0.f;
  __syncthreads();

  unsigned Tk[8];
  #pragma unroll
  for (int r = 0; r < 8; ++r) Tk[r] = rowAct[r] ? prefix[rbase + r] : 0u;

  for (int tile = wave; tile < NPTS / 16; tile += 8) {
    const int pidx = tile * 16 + col;
    const float px = P[3 * pidx + 0];
    const float py = P[3 * pidx + 1];
    const float pz = P[3 * pidx + 2];
    const float pn2 = fmaf(px, px, fmaf(py, py, pz * pz));
    v2f bfrag;
    if (hlf == 0) { bfrag.x = px; bfrag.y = py;  }
    else          { bfrag.x = pz; bfrag.y = pn2; }
    v8f acc = {};
    acc = __builtin_amdgcn_wmma_f32_16x16x4_f32(false, afrag, false, bfrag,
                                                (short)0, acc, false, false);
    #pragma unroll
    for (int r = 0; r < 8; ++r) {
      if (rowAct[r]) {
        const int row = rbase + r;
        const unsigned key = okey(acc[r]);
        if (key < Tk[r]) {
          atomicAdd(&sums[row][0], px);
          atomicAdd(&sums[row][1], py);
          atomicAdd(&sums[row][2], pz);
          if (mode) {
            atomicAdd(&sums[row][3], px * px);
            atomicAdd(&sums[row][4], py * py);
            atomicAdd(&sums[row][5], pz * pz);
            atomicAdd(&sums[row][6], px * py);
            atomicAdd(&sums[row][7], px * pz);
            atomicAdd(&sums[row][8], py * pz);
          }
        } else if (key == Tk[r]) {
          const unsigned pos = atomicAdd(&tiecnt[row], 1u);
          if (pos < 16u) tielist[row][pos] = (unsigned)pidx;
        }
      }
    }
  }
  __syncthreads();

  if (tid < nq) {
    unsigned cnt = tiecnt[tid]; if (cnt > 16u) cnt = 16u;
    // ascending insertion sort: lowest-index-first tie-break (matches top_k)
    for (unsigned i = 1; i < cnt; ++i) {
      const unsigned v = tielist[tid][i]; int j = (int)i - 1;
      while (j >= 0 && tielist[tid][j] > v) { tielist[tid][j + 1] = tielist[tid][j]; --j; }
      tielist[tid][j + 1] = v;
    }
    unsigned need = kth[tid]; if (need > cnt) need = cnt;
    float s[9];
    #pragma unroll
    for (int c = 0; c < 9; ++c) s[c] = sums[tid][c];
    for (unsigned i = 0; i < need; ++i) {
      const unsigned idx = tielist[tid][i];
      const float px = P[3 * idx + 0], py = P[3 * idx + 1], pz = P[3 * idx + 2];
      s[0] += px; s[1] += py; s[2] += pz;
      if (mode) {
        s[3] += px * px; s[4] += py * py; s[5] += pz * pz;
        s[6] += px * py; s[7] += px * pz; s[8] += py * pz;
      }
    }
    const int G = b * GN + qtile * 16 + tid;
    if (mode == 0) {
      outv[3 * G + 0] = s[0] * (1.0f / GS);
      outv[3 * G + 1] = s[1] * (1.0f / GS);
      outv[3 * G + 2] = s[2] * (1.0f / GS);
    } else {
      for (int c = 0; c < 9; ++c) outv[9 * G + c] = s[c];
    }
  }
}

__device__ __forceinline__ void cross3(const float* u, const float* v, float* w) {
  w[0] = u[1] * v[2] - u[2] * v[1];
  w[1] = u[2] * v[0] - u[0] * v[2];
  w[2] = u[0] * v[1] - u[1] * v[0];
}

// 240 analytic 3x3 symmetric eigensolves + nearest-group smoothness -> scalar
__global__ __launch_bounds__(256)
void finalize_kernel(const float* __restrict__ means,
                     const float* __restrict__ sums2,
                     float* __restrict__ out)
{
  __shared__ float dirx[240], diry[240], dirz[240];
  __shared__ float mxs[240], mys[240], mzs[240];
  __shared__ float elong[240], dlo[240];
  const int t = threadIdx.x;

  if (t < 240) {
    const float* s = sums2 + 9 * t;
    const float mux = s[0] * 0.01f, muy = s[1] * 0.01f, muz = s[2] * 0.01f;
    const float a00 = s[3] * 0.01f - mux * mux;
    const float a11 = s[4] * 0.01f - muy * muy;
    const float a22 = s[5] * 0.01f - muz * muz;
    const float a01 = s[6] * 0.01f - mux * muy;
    const float a02 = s[7] * 0.01f - mux * muz;
    const float a12 = s[8] * 0.01f - muy * muz;

    // trigonometric eigenvalues of symmetric 3x3
    const float q  = (a00 + a11 + a22) * (1.0f / 3.0f);
    const float p1 = a01 * a01 + a02 * a02 + a12 * a12;
    const float d0 = a00 - q, d1 = a11 - q, d2 = a22 - q;
    const float p2 = d0 * d0 + d1 * d1 + d2 * d2 + 2.0f * p1;
    const float p  = sqrtf(p2 * (1.0f / 6.0f)) + 1e-20f;
    const float ip = 1.0f / p;
    const float b00 = d0 * ip, b11 = d1 * ip, b22 = d2 * ip;
    const float b01 = a01 * ip, b02 = a02 * ip, b12 = a12 * ip;
    float r = 0.5f * (b00 * (b11 * b22 - b12 * b12)
                    - b01 * (b01 * b22 - b12 * b02)
                    + b02 * (b01 * b12 - b11 * b02));
    r = fminf(1.0f, fmaxf(-1.0f, r));
    const float phi  = acosf(r) * (1.0f / 3.0f);
    const float lmax = q + 2.0f * p * cosf(phi);
    const float lmin = q + 2.0f * p * cosf(phi + 2.09439510239319549f);
    const float lmid = 3.0f * q - lmax - lmin;

    elong[t] = (lmax - lmid) / (lmax + lmid + lmin + 1e-9f);

    // eigenvector of lmax via best cross product of rows of (A - lmax I)
    const float m00 = a00 - lmax, m11 = a11 - lmax, m22 = a22 - lmax;
    const float r0[3] = { m00, a01, a02 };
    const float r1[3] = { a01, m11, a12 };
    const float r2[3] = { a02, a12, m22 };
    float c01[3], c02[3], c12[3];
    cross3(r0, r1, c01); cross3(r0, r2, c02); cross3(r1, r2, c12);
    const float n01 = c01[0]*c01[0] + c01[1]*c01[1] + c01[2]*c01[2];
    const float n02 = c02[0]*c02[0] + c02[1]*c02[1] + c02[2]*c02[2];
    const float n12 = c12[0]*c12[0] + c12[1]*c12[1] + c12[2]*c12[2];
    const float* best = c01; float nb = n01;
    if (n02 > nb) { best = c02; nb = n02; }
    if (n12 > nb) { best = c12; nb = n12; }
    const float inv = rsqrtf(nb + 1e-30f);
    dirx[t] = best[0] * inv; diry[t] = best[1] * inv; dirz[t] = best[2] * inv;

    mxs[t] = means[3 * t + 0]; mys[t] = means[3 * t + 1]; mzs[t] = means[3 * t + 2];
  }
  __syncthreads();

  if (t < 240) {
    const int b = t / GN, g = t % GN;
    const float bx = mxs[t], by = mys[t], bz = mzs[t];
    float bestd = 3.4e38f; int j = b * GN;
    for (int g2 = 0; g2 < GN; ++g2) {
      if (g2 == g) continue;
      const int u = b * GN + g2;
      const float dx = bx - mxs[u], dy = by - mys[u], dz = bz - mzs[u];
      const float dd = dx * dx + dy * dy + dz * dz;
      if (dd < bestd) { bestd = dd; j = u; }
    }
    const float cs = dirx[t] * dirx[j] + diry[t] * diry[j] + dirz[t] * dirz[j];
    dlo[t] = 1.0f - cs * cs;
  }
  __syncthreads();

  if (t == 0) {
    float E = 0.f, D = 0.f;
    for (int i = 0; i < 240; ++i) { E += elong[i]; D += dlo[i]; }
    out[0] = -E * (1.0f / 8.0f) + D * (1.0f / 240.0f);   // BETA = 1
  }
}

extern "C" void kernel_launch(void* const* d_in, const int* in_sizes, int n_in,
                              void* d_out, int out_size, void* d_ws, size_t ws_size,
                              hipStream_t stream)
{
  (void)in_sizes; (void)n_in; (void)out_size; (void)ws_size;
  const float* pts = (const float*)d_in[0];
  float* out = (float*)d_out;
  float* ws  = (float*)d_ws;

  float* means = ws;            // 8*30*3 = 720 floats
  float* sums2 = ws + 1024;     // 240*9  = 2160 floats

  dim3 grid(2, 8);              // 2 query tiles x 8 batches

  // Pass 1: queries = strided centers from the point cloud -> group means
  knn_select_kernel<<<grid, 256, 0, stream>>>(pts, pts, NPTS * 3, 3 * CSTRIDE, means, 0);
  // Pass 2: queries = group means -> Σx and Σxxᵀ over 100 NN
  knn_select_kernel<<<grid, 256, 0, stream>>>(pts, means, GN * 3, 3, sums2, 1);
  // Eigensolves + losses -> scalar
  finalize_kernel<<<1, 256, 0, stream>>>(means, sums2, out);
}